// CogitatDeepSetNorm_45363444580781
// MI455X (gfx1250) — compile-verified
//
#include <hip/hip_runtime.h>
#include <stdint.h>

// ---- problem constants (from reference) ----
#define CDN_N      131072   // rows
#define CDN_DIN    256      // input dims
#define CDN_SUBS   64       // segments
#define CDN_DMID   128      // middle dims (folds into a scalar factor)
#define CDN_DOUT   256      // output dims

// Phase-1 tiling
#define CDN_ROWS_PER_WAVE 16
#define CDN_CHUNK_COLS    32                 // columns per LDS chunk
#define CDN_NCHUNK        (CDN_DIN / CDN_CHUNK_COLS)   // 8
#define CDN_LDS_STRIDE    36                 // dwords per row (32 + 4 pad -> conflict-free)
#define CDN_CHUNK_DW      (CDN_ROWS_PER_WAVE * CDN_LDS_STRIDE)  // 576 dwords per buffer

typedef __attribute__((ext_vector_type(2))) float v2f;
typedef __attribute__((ext_vector_type(4))) float v4f;
typedef __attribute__((ext_vector_type(8))) float v8f;

// ---------------------------------------------------------------------------
// Phase 1: per-row sum of x via V_WMMA_F32_16X16X4_F32, fed through LDS by
// GLOBAL_LOAD_ASYNC_TO_LDS_B128 (ASYNCcnt-tracked DMA, coalesced 512B/instr).
//
// One wave handles 16 rows. Global tile is staged in a wave-private,
// double-buffered LDS region (no cross-wave sharing -> no barriers; ASYNCcnt
// is per-wave and completes in order, so s_wait_asynccnt 4 == "front chunk
// landed" while the next chunk's 4 DMAs are still in flight).
//
// WMMA operands: A = ones (16x4), B = 16-row x 4-col data tile read from LDS
// (B layout: lanes 0-15 hold K=0..1, lanes 16-31 hold K=2..3, N = lane&15).
// Chained C across 64 K-steps; afterwards c[0] lane l = rowsum(row (l&15)).
// ---------------------------------------------------------------------------
__global__ void cdn_rowsum_wmma(const float* __restrict__ x,
                                float* __restrict__ rws) {
  __shared__ float sb[8][2][CDN_CHUNK_DW];   // 8 waves x 2 buffers x 2304B = 36 KB

  const int lane    = threadIdx.x & 31;
  const int wave    = threadIdx.x >> 5;
  const int rowbase = blockIdx.x * 128 + wave * 16;   // 8 waves * 16 rows

  // ---- async-copy lane mapping: 4 rows x (8 lanes * 16B) per instruction ----
  const int cp_row = lane >> 3;          // 0..3   row within 4-row group
  const int cp_col = (lane & 7) * 4;     // 0..28  column (floats) within chunk

  // ---- wmma-read lane mapping ----
  const int rd_row  = lane & 15;                 // row within tile
  const int rd_koff = (lane < 16) ? 0 : 2;       // K offset within 4-col step

  const float* gbase = x + (size_t)rowbase * CDN_DIN;

  // LDS byte addresses: generic pointers to __shared__ carry the LDS aperture
  // in the high bits; the low 32 bits are the LDS byte offset (ISA 10.2).
  uint32_t lds0 = (uint32_t)(uintptr_t)&sb[wave][0][0];
  uint32_t lds1 = (uint32_t)(uintptr_t)&sb[wave][1][0];

  // Issue one chunk's 4 async b128 DMAs (16 rows x 32 cols = 2 KB).
  auto issue_chunk = [&](int c, uint32_t ldsbase) {
#pragma unroll
    for (int i = 0; i < 4; ++i) {
      const int  r = i * 4 + cp_row;                              // row in tile
      const float* g = gbase + (size_t)r * CDN_DIN + c * CDN_CHUNK_COLS + cp_col;
      uint32_t l = ldsbase + (uint32_t)(r * CDN_LDS_STRIDE + cp_col) * 4u;
      asm volatile("global_load_async_to_lds_b128 %0, %1, off"
                   :: "v"(l), "v"(g) : "memory");
    }
  };

  v2f a = {1.0f, 1.0f};   // ones matrix (A, 16x4)
  v8f c = {};             // accumulator (C/D, 16x16 f32)

  issue_chunk(0, lds0);

#pragma unroll
  for (int ch = 0; ch < CDN_NCHUNK; ++ch) {
    const uint32_t cur = (ch & 1) ? lds1 : lds0;
    if (ch + 1 < CDN_NCHUNK) {
      issue_chunk(ch + 1, (ch & 1) ? lds0 : lds1);
      asm volatile("s_wait_asynccnt 0x4" ::: "memory");  // front chunk done
    } else {
      asm volatile("s_wait_asynccnt 0x0" ::: "memory");
    }

    // Consume chunk: 8 K-steps of 4 columns. Bank-conflict-free ds_load_b64:
    // dword addr = 36*rd_row + koff + 4k  (36*l mod 64 distinct for l=0..15).
    const float* lp = (const float*)(uintptr_t)(uint64_t)
        (0);  // placeholder, replaced below via shared-array indexing
    (void)lp;
    const float* pr = &sb[wave][ch & 1][rd_row * CDN_LDS_STRIDE + rd_koff];
#pragma unroll
    for (int k = 0; k < CDN_CHUNK_COLS / 4; ++k) {
      v2f b = *(const v2f*)(pr + k * 4);
      // 8 args: (neg_a, A, neg_b, B, c_mod, C, reuse_a, reuse_b)
      c = __builtin_amdgcn_wmma_f32_16x16x4_f32(
          false, a, false, b, (short)0, c, false, false);
    }
  }

  if (lane < 16) rws[rowbase + lane] = c[0];
}

// ---------------------------------------------------------------------------
// Phase 2: deterministic segment reduction. One block per segment; each
// thread accumulates a fixed strided subset, then a fixed LDS tree reduce.
// rws + sub stay resident in the 192MB L2 across the 64 blocks.
// ---------------------------------------------------------------------------
__global__ void cdn_seg_reduce(const float* __restrict__ rws,
                               const int* __restrict__ sub,
                               float* __restrict__ segsum,
                               int* __restrict__ segcnt) {
  __shared__ float ssum[256];
  __shared__ int   scnt[256];
  const int seg = blockIdx.x;
  const int t   = threadIdx.x;

  float acc = 0.0f;
  int   cnt = 0;
  for (int i = t; i < CDN_N; i += 256) {
    if (sub[i] == seg) { acc += rws[i]; cnt++; }
  }
  ssum[t] = acc;
  scnt[t] = cnt;
  __syncthreads();
  for (int off = 128; off > 0; off >>= 1) {
    if (t < off) { ssum[t] += ssum[t + off]; scnt[t] += scnt[t + off]; }
    __syncthreads();
  }
  if (t == 0) { segsum[seg] = ssum[0]; segcnt[seg] = scnt[0]; }
}

// ---------------------------------------------------------------------------
// Phase 3a: per-segment scalar. Empty segment falls back to rowsum(x[0])
// (reference substitutes x[0] for the mean). Pre-multiplies by D_MID so the
// output pass is a single fma.
// ---------------------------------------------------------------------------
__global__ void cdn_sval(const float* __restrict__ segsum,
                         const int* __restrict__ segcnt,
                         const float* __restrict__ rws,
                         const float* __restrict__ Gamma,
                         float* __restrict__ sval_scaled) {
  const int s = threadIdx.x;
  if (s < CDN_SUBS) {
    const int   c  = segcnt[s];
    const float mt = (c > 0) ? (segsum[s] / (float)c) : rws[0];
    const float sv = fmaxf(0.0f, Gamma[0] * mt);
    sval_scaled[s] = sv * (float)CDN_DMID;
  }
}

// ---------------------------------------------------------------------------
// Phase 3b: broadcast output. One wave per row: two coalesced b128 stores
// cover the contiguous 1KB row. Pure store-bandwidth pass (~128 MB).
// ---------------------------------------------------------------------------
__global__ void cdn_broadcast_out(const float* __restrict__ rws,
                                  const int* __restrict__ sub,
                                  const float* __restrict__ sval_scaled,
                                  const float* __restrict__ Lambda,
                                  float* __restrict__ out) {
  const int lane = threadIdx.x & 31;
  const int wave = threadIdx.x >> 5;
  const int row  = blockIdx.x * 8 + wave;

  const float o = fmaxf(0.0f, Lambda[0] * (rws[row] + sval_scaled[sub[row]]));
  v4f v = {o, o, o, o};

  float* dst = out + (size_t)row * CDN_DOUT;
  *(v4f*)(dst + lane * 4)       = v;   // columns   0..127
  *(v4f*)(dst + 128 + lane * 4) = v;   // columns 128..255
}

// ---------------------------------------------------------------------------
// Launch: d_in = { x[f32 N*256], sub[i32 N], Gamma[f32 1], Lambda[f32 1] }
// Workspace layout (floats): rws[N] | segsum[64] | segcnt[64] | sval[64]
// ---------------------------------------------------------------------------
extern "C" void kernel_launch(void* const* d_in, const int* in_sizes, int n_in,
                              void* d_out, int out_size, void* d_ws, size_t ws_size,
                              hipStream_t stream) {
  const float* x      = (const float*)d_in[0];
  const int*   sub    = (const int*)d_in[1];
  const float* Gamma  = (const float*)d_in[2];
  const float* Lambda = (const float*)d_in[3];
  float*       out    = (float*)d_out;

  float* rws         = (float*)d_ws;
  float* segsum      = rws + CDN_N;
  int*   segcnt      = (int*)(segsum + CDN_SUBS);
  float* sval_scaled = (float*)(segcnt + CDN_SUBS);

  cdn_rowsum_wmma  <<<CDN_N / 128, 256, 0, stream>>>(x, rws);
  cdn_seg_reduce   <<<CDN_SUBS,    256, 0, stream>>>(rws, sub, segsum, segcnt);
  cdn_sval         <<<1,            64, 0, stream>>>(segsum, segcnt, rws, Gamma, sval_scaled);
  cdn_broadcast_out<<<CDN_N / 8,   256, 0, stream>>>(rws, sub, sval_scaled, Lambda, out);
}